// Decoder_36764920054393
// MI455X (gfx1250) — compile-verified
//
#include <hip/hip_runtime.h>
#include <hip/hip_bf16.h>

#define V_ 512
#define E_ 256
#define H_ 256
#define EMB_ 256
#define B_ 64
#define S_ 4096
#define SE_ (S_ * E_)          // 1,048,576
#define G3H_ (3 * H_)          // 768
#define NCHUNK 16
#define CHUNK_S (S_ / NCHUNK)  // 256

typedef __attribute__((ext_vector_type(2))) float v2f;
typedef __attribute__((ext_vector_type(8))) float v8f;

// ---------------------------------------------------------------------------
// K0: hw[b] = dot(h[b,:], attn_w[0:H]) ; 64 blocks x 256 threads
// ---------------------------------------------------------------------------
__global__ void hw_kernel(const float* __restrict__ h,
                          const float* __restrict__ attn_w,
                          float* __restrict__ hw) {
  __shared__ float red[256];
  const int b = blockIdx.x, tid = threadIdx.x;
  red[tid] = h[b * H_ + tid] * attn_w[tid];
  __syncthreads();
  for (int off = 128; off; off >>= 1) {
    if (tid < off) red[tid] += red[tid + off];
    __syncthreads();
  }
  if (tid == 0) hw[b] = red[0];
}

// ---------------------------------------------------------------------------
// K1: scores[s,b] = dot(enc[b,s,:], w_e) + hw[b] + attn_b ; softmax over b;
//     write directly into the aw output segment at s*64+b (flat identity).
//     One block per s. 8 waves (wave32), each wave handles 8 b values.
// ---------------------------------------------------------------------------
__global__ void scores_softmax_kernel(const float* __restrict__ enc,
                                      const float* __restrict__ attn_w,
                                      const float* __restrict__ attn_b,
                                      const float* __restrict__ hw,
                                      float* __restrict__ aw_out) {
  __shared__ float we[E_];
  __shared__ float sc[B_];
  __shared__ float red[64];
  const int s = blockIdx.x;
  const int tid = threadIdx.x;
  const int wave = tid >> 5;
  const int lane = tid & 31;

  we[tid] = attn_w[H_ + tid];  // w_e
  __syncthreads();

  // each wave: 8 batches; lane covers 8 contiguous floats of the 256-dot
  const float4* wev = reinterpret_cast<const float4*>(&we[lane * 8]);
  const float4 w0 = wev[0], w1 = wev[1];
  for (int i = 0; i < 8; ++i) {
    const int b = wave * 8 + i;
    const float4* ev = reinterpret_cast<const float4*>(
        enc + (size_t)b * SE_ + (size_t)s * E_ + lane * 8);
    const float4 e0 = ev[0], e1 = ev[1];
    float v = e0.x * w0.x + e0.y * w0.y + e0.z * w0.z + e0.w * w0.w +
              e1.x * w1.x + e1.y * w1.y + e1.z * w1.z + e1.w * w1.w;
    for (int off = 16; off; off >>= 1) v += __shfl_xor(v, off, 32);
    if (lane == 0) sc[b] = v + hw[b] + attn_b[0];
  }
  __syncthreads();

  // softmax over the 64 batch entries of this s-row
  if (tid < 64) red[tid] = sc[tid];
  __syncthreads();
  for (int off = 32; off; off >>= 1) {
    if (tid < off) red[tid] = fmaxf(red[tid], red[tid + off]);
    __syncthreads();
  }
  const float mx = red[0];
  __syncthreads();
  if (tid < 64) {
    const float e = __expf(sc[tid] - mx);
    sc[tid] = e;
    red[tid] = e;
  }
  __syncthreads();
  for (int off = 32; off; off >>= 1) {
    if (tid < off) red[tid] += red[tid + off];
    __syncthreads();
  }
  const float inv = 1.0f / red[0];
  if (tid < 64) aw_out[(size_t)s * 64 + tid] = sc[tid] * inv;
}

// ---------------------------------------------------------------------------
// K2: ctx partials: ctx_part[chunk][b][e] = sum_{s in chunk} aw[b,s]*enc[b,s,e]
//     grid (NCHUNK, B_), 256 threads.
//     b128 per-lane loads: thread = (row 0..3, col4 0..63); each thread owns
//     4 consecutive e-columns (float4) and strides over s with step 4;
//     4 row-partials are combined through LDS at the end.
// ---------------------------------------------------------------------------
__global__ void ctx_partial_kernel(const float* __restrict__ enc,
                                   const float* __restrict__ aw,  // flat (B,S)
                                   float* __restrict__ ctx_part) {
  __shared__ float awl[CHUNK_S];
  __shared__ float part[4][E_];
  const int chunk = blockIdx.x;
  const int b = blockIdx.y;
  const int tid = threadIdx.x;
  const int row = tid >> 6;         // 0..3  (s % 4 stripe)
  const int c4 = (tid & 63) * 4;    // e column group (4 floats)

  awl[tid] = aw[(size_t)b * S_ + chunk * CHUNK_S + tid];
  __syncthreads();

  const float* encb = enc + (size_t)b * SE_ + (size_t)(chunk * CHUNK_S) * E_;
  float4 acc = make_float4(0.f, 0.f, 0.f, 0.f);
  for (int i = row; i < CHUNK_S; i += 4) {
    if (i + 16 < CHUNK_S)
      __builtin_prefetch(&encb[(size_t)(i + 16) * E_ + c4], 0, 0);
    const float4 ev =
        *reinterpret_cast<const float4*>(&encb[(size_t)i * E_ + c4]);
    const float w = awl[i];
    acc.x += w * ev.x;
    acc.y += w * ev.y;
    acc.z += w * ev.z;
    acc.w += w * ev.w;
  }
  part[row][c4 + 0] = acc.x;
  part[row][c4 + 1] = acc.y;
  part[row][c4 + 2] = acc.z;
  part[row][c4 + 3] = acc.w;
  __syncthreads();

  // combine the 4 s-stripes; thread tid owns column e = tid
  const float v = part[0][tid] + part[1][tid] + part[2][tid] + part[3][tid];
  ctx_part[((size_t)chunk * B_ + b) * E_ + tid] = v;
}

// ---------------------------------------------------------------------------
// K3: reduce partials; x = relu([ctx, emb[dec[b]]]) ; 64 blocks x 256 threads
// ---------------------------------------------------------------------------
__global__ void build_x_kernel(const float* __restrict__ ctx_part,
                               const float* __restrict__ emb,
                               const int* __restrict__ dec,
                               float* __restrict__ x) {
  const int b = blockIdx.x, e = threadIdx.x;
  float acc = 0.0f;
  for (int c = 0; c < NCHUNK; ++c)
    acc += ctx_part[((size_t)c * B_ + b) * E_ + e];
  x[(size_t)b * (E_ + EMB_) + e] = fmaxf(acc, 0.0f);
  const float xe = emb[(size_t)dec[b] * EMB_ + e];
  x[(size_t)b * (E_ + EMB_) + E_ + e] = fmaxf(xe, 0.0f);
}

// ---------------------------------------------------------------------------
// K4: WMMA f32 GEMM:  C[m,n] = sum_k A[m,k]*Bw[n,k] + bias[n]
//     A: row-major M x K ; Bw: row-major N x K (i.e. computes A @ Bw^T).
//     One wave (32 threads) per 16x16 C tile using V_WMMA_F32_16X16X4_F32.
// ---------------------------------------------------------------------------
__global__ void wmma_gemm_nt(const float* __restrict__ A,
                             const float* __restrict__ Bw,
                             const float* __restrict__ bias,
                             float* __restrict__ C,
                             int K, int ldc) {
  const int n0 = blockIdx.x * 16;
  const int m0 = blockIdx.y * 16;
  const int lane = threadIdx.x;      // 0..31 (wave32)
  const int half = lane >> 4;        // 0: K pair {0,1} ; 1: K pair {2,3}
  const int l = lane & 15;

  v8f acc = {};
  for (int k0 = 0; k0 < K; k0 += 4) {
    const int ka = k0 + half * 2;
    v2f a, bfrag;
    // A-frag: lanes 0-15 hold A[m0+l][k0..k0+1], lanes 16-31 hold K=k0+2..k0+3
    a.x = A[(size_t)(m0 + l) * K + ka];
    a.y = A[(size_t)(m0 + l) * K + ka + 1];
    // B-frag: B[k][n] == Bw[n][k]; mirrored striping over lanes
    bfrag.x = Bw[(size_t)(n0 + l) * K + ka];
    bfrag.y = Bw[(size_t)(n0 + l) * K + ka + 1];
    acc = __builtin_amdgcn_wmma_f32_16x16x4_f32(
        /*neg_a=*/false, a, /*neg_b=*/false, bfrag,
        /*c_mod=*/(short)0, acc, /*reuse_a=*/false, /*reuse_b=*/false);
  }
  // C/D layout: VGPR r -> (M=m0+r, N=n0+lane) for lanes 0-15,
  //             (M=m0+8+r, N=n0+lane-16) for lanes 16-31
  const int mrow = m0 + half * 8;
  const int ncol = n0 + l;
  const float bv = bias[ncol];
#pragma unroll
  for (int r = 0; r < 8; ++r)
    C[(size_t)(mrow + r) * ldc + ncol] = acc[r] + bv;
}

// ---------------------------------------------------------------------------
// K5: GRU pointwise -> h_new ; 64 blocks x 256 threads
// ---------------------------------------------------------------------------
__global__ void gru_kernel(const float* __restrict__ gi,
                           const float* __restrict__ gh,
                           const float* __restrict__ h,
                           float* __restrict__ hnew) {
  const int b = blockIdx.x, j = threadIdx.x;
  const size_t o = (size_t)b * G3H_;
  const float ir = gi[o + j],         hr = gh[o + j];
  const float iz = gi[o + H_ + j],    hz = gh[o + H_ + j];
  const float in_ = gi[o + 2*H_ + j], hn = gh[o + 2*H_ + j];
  const float r = 1.0f / (1.0f + __expf(-(ir + hr)));
  const float z = 1.0f / (1.0f + __expf(-(iz + hz)));
  const float n = tanhf(in_ + r * hn);
  const float hv = h[(size_t)b * H_ + j];
  hnew[(size_t)b * H_ + j] = (1.0f - z) * n + z * hv;
}

// ---------------------------------------------------------------------------
// K6: row-wise log_softmax over V=512 ; 64 blocks x 256 threads
// ---------------------------------------------------------------------------
__global__ void logsoftmax_kernel(const float* __restrict__ lp,
                                  float* __restrict__ out) {
  __shared__ float red[256];
  const int b = blockIdx.x, tid = threadIdx.x;
  const float v0 = lp[(size_t)b * V_ + tid];
  const float v1 = lp[(size_t)b * V_ + 256 + tid];
  red[tid] = fmaxf(v0, v1);
  __syncthreads();
  for (int off = 128; off; off >>= 1) {
    if (tid < off) red[tid] = fmaxf(red[tid], red[tid + off]);
    __syncthreads();
  }
  const float mx = red[0];
  __syncthreads();
  red[tid] = __expf(v0 - mx) + __expf(v1 - mx);
  __syncthreads();
  for (int off = 128; off; off >>= 1) {
    if (tid < off) red[tid] += red[tid + off];
    __syncthreads();
  }
  const float lse = mx + __logf(red[0]);
  out[(size_t)b * V_ + tid] = v0 - lse;
  out[(size_t)b * V_ + 256 + tid] = v1 - lse;
}

// ---------------------------------------------------------------------------
extern "C" void kernel_launch(void* const* d_in, const int* in_sizes, int n_in,
                              void* d_out, int out_size, void* d_ws, size_t ws_size,
                              hipStream_t stream) {
  const int*   dec     = (const int*)d_in[0];
  const float* h       = (const float*)d_in[1];
  const float* enc     = (const float*)d_in[2];
  const float* emb     = (const float*)d_in[3];
  const float* attn_w  = (const float*)d_in[4];
  const float* attn_b  = (const float*)d_in[5];
  const float* w_ih    = (const float*)d_in[6];
  const float* w_hh    = (const float*)d_in[7];
  const float* b_ih    = (const float*)d_in[8];
  const float* b_hh    = (const float*)d_in[9];
  const float* dense_w = (const float*)d_in[10];
  const float* dense_b = (const float*)d_in[11];

  float* out        = (float*)d_out;
  float* logits_out = out;                     // 64*512  = 32768
  float* hnew_out   = out + B_ * V_;           // 64*256  = 16384
  float* aw_out     = out + B_ * V_ + B_ * H_; // 64*4096 = 262144

  float* ws       = (float*)d_ws;
  float* hw       = ws;                                   // 64
  float* ctx_part = hw + B_;                              // 16*64*256
  float* x        = ctx_part + (size_t)NCHUNK * B_ * E_;  // 64*512
  float* gi       = x + (size_t)B_ * (E_ + EMB_);         // 64*768
  float* gh       = gi + (size_t)B_ * G3H_;               // 64*768
  float* lp       = gh + (size_t)B_ * G3H_;               // 64*512

  hipLaunchKernelGGL(hw_kernel, dim3(B_), dim3(256), 0, stream, h, attn_w, hw);
  hipLaunchKernelGGL(scores_softmax_kernel, dim3(S_), dim3(256), 0, stream,
                     enc, attn_w, attn_b, hw, aw_out);
  hipLaunchKernelGGL(ctx_partial_kernel, dim3(NCHUNK, B_), dim3(256), 0, stream,
                     enc, aw_out, ctx_part);
  hipLaunchKernelGGL(build_x_kernel, dim3(B_), dim3(256), 0, stream,
                     ctx_part, emb, dec, x);
  // gi = x @ w_ih^T + b_ih   (M=64, N=768, K=512)
  hipLaunchKernelGGL(wmma_gemm_nt, dim3(G3H_ / 16, B_ / 16), dim3(32), 0, stream,
                     x, w_ih, b_ih, gi, E_ + EMB_, G3H_);
  // gh = h @ w_hh^T + b_hh   (M=64, N=768, K=256)
  hipLaunchKernelGGL(wmma_gemm_nt, dim3(G3H_ / 16, B_ / 16), dim3(32), 0, stream,
                     h, w_hh, b_hh, gh, H_, G3H_);
  hipLaunchKernelGGL(gru_kernel, dim3(B_), dim3(256), 0, stream, gi, gh, h, hnew_out);
  // logits_pre = h_new @ dense_w^T + dense_b  (M=64, N=512, K=256)
  hipLaunchKernelGGL(wmma_gemm_nt, dim3(V_ / 16, B_ / 16), dim3(32), 0, stream,
                     hnew_out, dense_w, dense_b, lp, H_, V_);
  hipLaunchKernelGGL(logsoftmax_kernel, dim3(B_), dim3(256), 0, stream, lp, logits_out);
}